// PredictiveReasoningBlock_42236708389368
// MI455X (gfx1250) — compile-verified
//
#include <hip/hip_runtime.h>
#include <hip/hip_bf16.h>

// MI455X / gfx1250 implementation.
// FLOP-dominant 3x3 convs run as implicit GEMM on V_WMMA_F32_16X16X4_F32
// (exact fp32, M=oc, N=225 spatial, K=IC*9). Front section (LN, cal errors,
// gates, MLP) is a fused per-batch LDS kernel.
// Round-2 changes: im2col k->LDS-offset LUT (kills per-iter div/mul address
// math), dropped redundant out-of-range cndmask guards (dead-lane columns are
// discarded at store; LDS padded so speculative addresses stay in-bounds),
// K-loop unrolled x4.

#define EPS 1e-5f
#define BATCH 2048
#define CCH 25      // C
#define TT 9        // T
#define LL 25       // L
#define SP 225      // T*L spatial
#define MICH 100    // MI

typedef __attribute__((ext_vector_type(2))) float v2f;
typedef __attribute__((ext_vector_type(8))) float v8f;

// ---------------------------------------------------------------- helpers ---
__device__ __forceinline__ float bn_eval(float x, float g, float b, float m, float v) {
    return (x - m) * (g * rsqrtf(v + EPS)) + b;
}
__device__ __forceinline__ float gelu_exact(float x) {
    return 0.5f * x * (1.0f + erff(x * 0.70710678118654752440f));
}
__device__ __forceinline__ float sigmoidf(float x) {
    return 1.0f / (1.0f + __expf(-x));
}
__device__ __forceinline__ v8f wmma_f32(v2f a, v2f b, v8f c) {
    // D = A(16x4) * B(4x16) + C(16x16), exact fp32
    return __builtin_amdgcn_wmma_f32_16x16x4_f32(
        /*neg_a=*/false, a, /*neg_b=*/false, b,
        /*c_mod=*/(short)0, c, /*reuse_a=*/false, /*reuse_b=*/false);
}

// error-map tables: which xn t-slices feed each of the 7 error maps
__constant__ int kErrTA[7] = {0, 3, 6, 0, 3, 6, 2};
__constant__ int kErrTB[7] = {1, 4, 7, 2, 5, 8, 1};
__constant__ int kErrTG[7] = {2, 5, 8, 1, 4, 7, 0};
__constant__ int kErrP [7] = {0, 0, 0, 1, 1, 1, 2};   // cal1/cal2/cal3
// p1/p2/p3 row sources: >=0 -> xn slice t, -1 -> error map (kPE)
__constant__ int kPS[3][9] = {
    { 0, 1,-1, 3, 4,-1, 6, 7,-1},
    { 0,-1, 2, 3,-1, 5, 6,-1, 8},
    {-1, 1, 2,-1, 4, 5,-1, 7, 8}};
__constant__ int kPE[3][9] = {
    {-1,-1, 0,-1,-1, 1,-1,-1, 2},
    {-1, 3,-1,-1, 4,-1,-1, 5,-1},
    { 6,-1,-1, 4,-1,-1, 5,-1,-1}};

struct FrontParams {
    const float* x;
    const float* calw[3];
    const float* calg[3]; const float* calb[3];
    const float* calm[3]; const float* calv[3];
    const float* mlpw; const float* mlpb;
    const float* mlpg; const float* mlpbb; const float* mlpm; const float* mlpv;
    const float* mw[3]; const float* mbias[3];
    const float* mg[3]; const float* mb2[3]; const float* mm[3]; const float* mv[3];
    const float* lng; const float* lnb;
    float* pout;
};

// ------------------------------------------------------------ front kernel ---
// One block per batch. LDS: xn[25][9][25] + err[7][25][25] = 40 KB.
__global__ __launch_bounds__(256)
void front_kernel(FrontParams P) {
    __shared__ float sxn[CCH * SP];      // [c][t][l]
    __shared__ float serr[7 * 625];      // [map][o][l]
    const int b = blockIdx.x;
    const int tid = threadIdx.x;

    // load x[b]
    const float* xb = P.x + (size_t)b * (CCH * SP);
    for (int i = tid; i < CCH * SP; i += 256) sxn[i] = xb[i];
    __syncthreads();

    // LayerNorm over L per (c,t) row
    for (int r = tid; r < CCH * TT; r += 256) {
        float* row = &sxn[r * LL];
        float mu = 0.f;
        #pragma unroll
        for (int l = 0; l < LL; ++l) mu += row[l];
        mu *= (1.0f / LL);
        float var = 0.f;
        #pragma unroll
        for (int l = 0; l < LL; ++l) { float d = row[l] - mu; var += d * d; }
        var *= (1.0f / LL);
        float inv = rsqrtf(var + EPS);
        #pragma unroll
        for (int l = 0; l < LL; ++l)
            row[l] = (row[l] - mu) * inv * P.lng[l] + P.lnb[l];
    }
    __syncthreads();

    // 7 error maps: err[map][o][l] = gelu(xn[o][tG][l]) - relu(bn(convmix))
    for (int i = tid; i < 7 * 625; i += 256) {
        const int map = i / 625;
        const int o = (i % 625) / LL;
        const int l = i % LL;
        const int ps = kErrP[map];
        const int tA = kErrTA[map], tB = kErrTB[map], tG = kErrTG[map];
        const float* w = P.calw[ps];
        float s = 0.f;
        #pragma unroll
        for (int ic = 0; ic < CCH; ++ic) {
            s += w[o * 50 + ic * 2 + 0] * sxn[ic * SP + tA * LL + l];
            s += w[o * 50 + ic * 2 + 1] * sxn[ic * SP + tB * LL + l];
        }
        float y = bn_eval(s, P.calg[ps][o], P.calb[ps][o], P.calm[ps][o], P.calv[ps][o]);
        y = fmaxf(y, 0.f);
        serr[i] = gelu_exact(sxn[o * SP + tG * LL + l]) - y;
    }
    __syncthreads();

    // per-(c,t) row: gates m1/m2/m3, mix, MLP(25x25)+BN+ReLU -> p
    for (int r = tid; r < CCH * TT; r += 256) {
        const int c = r / TT;
        const int t = r % TT;
        float pr[3][LL];
        #pragma unroll
        for (int v = 0; v < 3; ++v) {
            const int st = kPS[v][t];
            if (st >= 0) {
                #pragma unroll
                for (int l = 0; l < LL; ++l) pr[v][l] = sxn[c * SP + st * LL + l];
            } else {
                const int e = kPE[v][t];
                #pragma unroll
                for (int l = 0; l < LL; ++l) pr[v][l] = serr[e * 625 + c * LL + l];
            }
        }
        float gk[3];
        #pragma unroll
        for (int k = 0; k < 3; ++k) {
            const float* w = P.mw[k];
            float s = P.mbias[k][0];
            #pragma unroll
            for (int l = 0; l < LL; ++l) {
                s += w[l] * pr[0][l];
                s += w[LL + l] * pr[1][l];
                s += w[2 * LL + l] * pr[2][l];
            }
            float y = bn_eval(s, P.mg[k][0], P.mb2[k][0], P.mm[k][0], P.mv[k][0]);
            gk[k] = sigmoidf(fmaxf(y, 0.f));
        }
        float sr[LL];
        #pragma unroll
        for (int l = 0; l < LL; ++l)
            sr[l] = gk[0] * pr[0][l] + gk[1] * pr[1][l] + gk[2] * pr[2][l];

        float* outrow = P.pout + (size_t)b * (CCH * SP) + c * SP + t * LL;
        #pragma unroll
        for (int o = 0; o < LL; ++o) {
            float a = P.mlpb[o];
            #pragma unroll
            for (int l = 0; l < LL; ++l) a += sr[l] * P.mlpw[o * LL + l];
            float y = bn_eval(a, P.mlpg[o], P.mlpbb[o], P.mlpm[o], P.mlpv[o]);
            outrow[o] = fmaxf(y, 0.f);
        }
    }
}

// ---------------------------------------------------------- conv3x3 kernel ---
// Implicit GEMM on WMMA fp32 16x16x4. grid = (B, ceil(OC/16)). 256 thr = 8 waves,
// each wave owns 2 N-tiles (16 spatial columns). K = IC*9, chunked by 25 input
// channels; input chunk staged in LDS with zero halo [26][11][27] (ic row 25 is
// zeroed so the K tail 225..227 reads zeros). k -> LDS offset mapping comes from
// a 228-entry LUT; out-of-range columns compute garbage into lanes whose D
// columns are never stored (LDS padded so their addresses stay in-bounds).
template <int IC, int OC, bool ADD_SC>
__global__ __launch_bounds__(256)
void conv3x3_bn_relu(const float* __restrict__ in, const float* __restrict__ w,
                     const float* __restrict__ bg, const float* __restrict__ bb,
                     const float* __restrict__ bm, const float* __restrict__ bv,
                     const float* __restrict__ sc, float* __restrict__ out) {
    constexpr int NCHUNK = IC / 25;     // 1 for conv1, 4 for conv2
    constexpr int KW = IC * 9;          // weight K stride
    __shared__ float s_in[26 * 297 + 64];  // [ic 0..25][t+1 0..10][l+1 0..26] + slack
    __shared__ float s_wt[16 * 228];       // [oc_local][k], zero padded
    __shared__ int   s_off[228];           // k -> ic*297 + kh*27 + kw

    const int b = blockIdx.x;
    const int mt = blockIdx.y;
    const int tid = threadIdx.x;
    const int wave = tid >> 5;
    const int lid = tid & 31;
    const int mrow = lid & 15;                 // A: oc row / B,D: column
    const int khi = (lid < 16) ? 0 : 2;        // fp32 A/B K sub-lane offset
    const int n0 = (wave) * 16 + mrow;         // spatial col, tile 0
    const int n1 = (wave + 8) * 16 + mrow;     // spatial col, tile 1

    // k -> LDS element-offset LUT (covers the zero tail k=225..227: ic=25 row)
    for (int i = tid; i < 228; i += 256) {
        const int ic = i / 9;
        const int rr = i - ic * 9;
        const int kh = rr / 3;
        s_off[i] = ic * 297 + kh * 27 + (rr - kh * 3);
    }

    // per-tile base pointers into the halo'd input (valid for n up to 255)
    const int t0 = n0 / LL, l0 = n0 - t0 * LL;
    const int t1 = n1 / LL, l1 = n1 - t1 * LL;
    const float* bp0 = s_in + t0 * 27 + l0;
    const float* bp1 = s_in + t1 * 27 + l1;
    const float* ap  = s_wt + mrow * 228;

    v8f acc0 = {};
    v8f acc1 = {};

    for (int ch = 0; ch < NCHUNK; ++ch) {
        __syncthreads();
        const float* src = in + ((size_t)b * IC + ch * 25) * SP;
        if (ch + 1 < NCHUNK && lid == 0)
            __builtin_prefetch(src + 25 * SP, 0, 1);   // global_prefetch_b8
        for (int i = tid; i < 26 * 297; i += 256) {
            const int ic = i / 297;
            const int rem = i - ic * 297;
            const int t = rem / 27 - 1;
            const int l = rem - (t + 1) * 27 - 1;
            float val = 0.f;
            if (ic < 25 && t >= 0 && t < TT && l >= 0 && l < LL)
                val = src[ic * SP + t * LL + l];
            s_in[i] = val;
        }
        const float* wsrc = w + (size_t)(mt * 16) * KW + ch * SP;
        for (int i = tid; i < 16 * 228; i += 256) {
            const int oc = i / 228;
            const int k = i - oc * 228;
            float val = 0.f;
            if (mt * 16 + oc < OC && k < SP) val = wsrc[oc * KW + k];
            s_wt[i] = val;
        }
        __syncthreads();

        #pragma unroll 4
        for (int k0 = 0; k0 < 228; k0 += 4) {
            const int kA = k0 + khi;
            const int ox = s_off[kA];
            const int oy = s_off[kA + 1];
            v2f A;  A.x  = ap[kA];    A.y  = ap[kA + 1];   // contiguous -> ds_load_b64
            v2f B0; B0.x = bp0[ox];   B0.y = bp0[oy];
            v2f B1; B1.x = bp1[ox];   B1.y = bp1[oy];
            acc0 = wmma_f32(A, B0, acc0);
            acc1 = wmma_f32(A, B1, acc1);
        }
    }

    // store: D vgpr r -> oc = mt*16 + r + (lane<16?0:8), col = mrow
    const int ocb = mt * 16 + ((lid < 16) ? 0 : 8);
    #pragma unroll
    for (int r = 0; r < 8; ++r) {
        const int oc = ocb + r;
        if (oc >= OC) continue;
        const float scale = bg[oc] * rsqrtf(bv[oc] + EPS);
        const float shift = bb[oc] - bm[oc] * scale;
        const size_t base = ((size_t)b * OC + oc) * SP;
        if (n0 < SP) {
            float y = fmaxf(acc0[r] * scale + shift, 0.f);
            if (ADD_SC) y += sc[base + n0];
            out[base + n0] = y;
        }
        if (n1 < SP) {
            float y = fmaxf(acc1[r] * scale + shift, 0.f);
            if (ADD_SC) y += sc[base + n1];
            out[base + n1] = y;
        }
    }
}

// ------------------------------------------------------------------ launch ---
extern "C" void kernel_launch(void* const* d_in, const int* in_sizes, int n_in,
                              void* d_out, int out_size, void* d_ws, size_t ws_size,
                              hipStream_t stream) {
    (void)in_sizes; (void)n_in; (void)out_size; (void)ws_size;
    auto F = [&](int i) { return (const float*)d_in[i]; };

    // setup_inputs() flattening order:
    // 0:x  1-5:cal1{w,g,b,m,v}  6-10:cal2  11-15:cal3
    // 16:mlp.w 17:mlp.b 18-21:mlp.bn{g,b,m,v}
    // 22-27:m1{w,b,g,b,m,v} 28-33:m2 34-39:m3
    // 40:conv1.w 41-44:conv1.bn  45:conv2.w 46-49:conv2.bn  50:ln.g 51:ln.b
    float* p_ws = (float*)d_ws;                                   // [B,25,225]
    float* q_ws = p_ws + (size_t)BATCH * CCH * SP;                // [B,100,225]

    FrontParams P;
    P.x = F(0);
    for (int j = 0; j < 3; ++j) {
        const int base = 1 + 5 * j;
        P.calw[j] = F(base);     P.calg[j] = F(base + 1);
        P.calb[j] = F(base + 2); P.calm[j] = F(base + 3); P.calv[j] = F(base + 4);
    }
    P.mlpw = F(16); P.mlpb = F(17);
    P.mlpg = F(18); P.mlpbb = F(19); P.mlpm = F(20); P.mlpv = F(21);
    for (int k = 0; k < 3; ++k) {
        const int base = 22 + 6 * k;
        P.mw[k] = F(base);     P.mbias[k] = F(base + 1);
        P.mg[k] = F(base + 2); P.mb2[k] = F(base + 3);
        P.mm[k] = F(base + 4); P.mv[k] = F(base + 5);
    }
    P.lng = F(50); P.lnb = F(51);
    P.pout = p_ws;

    front_kernel<<<BATCH, 256, 0, stream>>>(P);

    conv3x3_bn_relu<CCH, MICH, false><<<dim3(BATCH, 7), 256, 0, stream>>>(
        p_ws, F(40), F(41), F(42), F(43), F(44), nullptr, q_ws);

    conv3x3_bn_relu<MICH, CCH, true><<<dim3(BATCH, 2), 256, 0, stream>>>(
        q_ws, F(45), F(46), F(47), F(48), F(49), F(0), (float*)d_out);
}